// SpecRadLoss_50680614092814
// MI455X (gfx1250) — compile-verified
//
#include <hip/hip_runtime.h>
#include <math.h>

// CDNA5 / gfx1250: wave32, WMMA f32 16x16x4.
typedef __attribute__((ext_vector_type(2))) float v2f;
typedef __attribute__((ext_vector_type(8))) float v8f;

#define L 1024
#define S 16
#define BATCH 32
#define PANEL (L * S)          // 16384 floats per batch panel
#define KCHUNK 512             // K rows staged in LDS per pass (2 passes)
#define N_ITERS 10

// ---------------------------------------------------------------------------
// Kernel 1: inverse column norms of sphere_points (L x S, row-major).
// ---------------------------------------------------------------------------
__global__ __launch_bounds__(256) void col_norms(const float* __restrict__ sp,
                                                 float* __restrict__ inv_norm) {
    __shared__ float red[256];
    const int s = blockIdx.x;
    float acc = 0.0f;
    for (int i = threadIdx.x; i < L; i += 256) {
        float v = sp[i * S + s];
        acc += v * v;
    }
    red[threadIdx.x] = acc;
    __syncthreads();
    for (int w = 128; w > 0; w >>= 1) {
        if ((int)threadIdx.x < w) red[threadIdx.x] += red[threadIdx.x + w];
        __syncthreads();
    }
    if (threadIdx.x == 0) inv_norm[s] = 1.0f / sqrtf(red[0]);
}

// ---------------------------------------------------------------------------
// Kernel 2: Y[b][i][s] = sp[i][s] * inv_norm[s]  (broadcast over batches).
// ---------------------------------------------------------------------------
__global__ __launch_bounds__(256) void init_y(const float* __restrict__ sp,
                                              const float* __restrict__ inv_norm,
                                              float* __restrict__ Y) {
    const int idx = blockIdx.x * 256 + threadIdx.x;   // < BATCH*PANEL
    const int ls  = idx & (PANEL - 1);                // i*S + s
    Y[idx] = sp[ls] * inv_norm[idx & (S - 1)];
}

// ---------------------------------------------------------------------------
// Kernel 3: structured-matrix panel GEMM via V_WMMA_F32_16X16X4_F32.
//   CIRC == false :  Out  = Toeplitz(coef) @ Bp      (T = A @ Y)
//   CIRC == true  :  Out -= Circulant(coef) @ Bp     (Y -= Ct @ T)
//
// Expanded-generator trick: stage h[2048] in LDS with
//   Toeplitz : h[p] = coef[|p-1023|]        => A[r][k] = h[1023 + k - r]
//   Circulant: h[p] = coef[(p-1023)&1023]   => A[r][k] = h[1023 + k - r]
// so both A-fragment elements (k, k+1) are ADJACENT LDS dwords with
// compile-time offsets after unrolling (-> ds_load_2addr, ~0 VALU/step).
//
// B panel is staged in LDS in K-chunks of 512 rows, pair-packed as v2f
// Bs2[k/2][n] = (B[k][n], B[k+1][n])  -> one 8B-aligned ds_load_b64 per WMMA.
//
// Grid: (8 tile-groups, 32 batches); block = 256 = 8 waves, 1 tile/wave.
// LDS: 8KB generator + 32KB panel chunk = 40KB -> 8 blocks/WGP.
// Fragment layouts per ISA 7.12.2 (f32 A 16x4: M = lane&15, lanes16-31 hold
// K+2; C/D: VGPR q = row q (+8 for upper half-wave), col = lane&15).
// ---------------------------------------------------------------------------
template <bool CIRC>
__global__ __launch_bounds__(256) void toeplitz_mm(const float* __restrict__ coef,
                                                   const float* __restrict__ Bp,
                                                   float* __restrict__ Op) {
    __shared__ float h[2 * L];                 // expanded generator, 8 KB
    __shared__ v2f   Bs2[KCHUNK * S / 2];      // pair-packed B chunk, 32 KB

    const int b    = blockIdx.y;
    const int tile = blockIdx.x * 8 + (threadIdx.x >> 5);
    const int lane = threadIdx.x & 31;

    // Stage expanded generator (abs/mod paid once here, not in the k-loop).
    for (int p = threadIdx.x; p < 2 * L; p += 256) {
        int d = p - (L - 1);
        int idx;
        if (CIRC) {
            idx = d & (L - 1);
        } else {
            idx = d < 0 ? -d : d;
            if (idx > L - 1) idx = L - 1;      // p == 2047 pad, value unused
        }
        h[p] = coef[b * L + idx];
    }

    const int r     = tile * 16 + (lane & 15); // output row (M = lane&15)
    const int khalf = (lane >> 4) * 2;         // K sub-pair select
    const int n     = lane & 15;               // output/B column
    const float* __restrict__ Bb = Bp + b * PANEL;

    v8f c = {};                                // f32 accumulator (8 VGPRs)

    for (int ch = 0; ch < L / KCHUNK; ++ch) {
        __syncthreads();                       // prev chunk fully consumed
        // Stage pair-packed B chunk: Bs2[k2*16+n] = (B[2k2][n], B[2k2+1][n])
        for (int idx = threadIdx.x; idx < KCHUNK * S / 2; idx += 256) {
            const int k2 = idx >> 4;
            const int nn = idx & 15;
            const int krow = ch * KCHUNK + 2 * k2;
            v2f v;
            v.x = Bb[krow * S + nn];
            v.y = Bb[krow * S + S + nn];
            Bs2[idx] = v;
        }
        __syncthreads();

        // All offsets below are compile-time constants after unrolling.
        const float* __restrict__ hA = h + (L - 1 - r + khalf + ch * KCHUNK);
        const v2f*   __restrict__ Bl = Bs2 + (khalf >> 1) * S + n;

        #pragma unroll 8
        for (int k0 = 0; k0 < KCHUNK; k0 += 4) {
            v2f a;
            a.x = hA[k0];                      // adjacent LDS dwords
            a.y = hA[k0 + 1];
            v2f bb = Bl[(k0 >> 1) * S];        // one ds_load_b64
            c = __builtin_amdgcn_wmma_f32_16x16x4_f32(false, a, false, bb,
                                                      (short)0, c, false, false);
        }
    }

    float* __restrict__ Ob = Op + b * PANEL;
    const int rowbase = tile * 16 + ((lane >> 4) << 3);
    #pragma unroll
    for (int q = 0; q < 8; ++q) {
        const int o = (rowbase + q) * S + n;
        if (CIRC) Ob[o] = Ob[o] - c[q];
        else      Ob[o] = c[q];
    }
}

// ---------------------------------------------------------------------------
// Kernel 4: per-batch value.  y += ALPHA*sp ; per-column sumsq over L;
// val_b = max_s (sumsq)^(1/20).  Fixed-order sums (deterministic).
// ---------------------------------------------------------------------------
__global__ __launch_bounds__(1024) void batch_val(const float* __restrict__ Y,
                                                  const float* __restrict__ sp,
                                                  const float* __restrict__ inv_norm,
                                                  float* __restrict__ vals) {
    __shared__ float part[1024];
    __shared__ float colv[S];
    const int b = blockIdx.x;
    const int s = threadIdx.x & (S - 1);
    const int g = threadIdx.x >> 4;            // 64 row-groups of 16
    const float* __restrict__ Yb = Y + b * PANEL;
    const float inv = inv_norm[s];

    float acc = 0.0f;
    #pragma unroll
    for (int ii = 0; ii < 16; ++ii) {
        const int i = g * 16 + ii;
        const float v = Yb[i * S + s] + 1e-12f * sp[i * S + s] * inv;
        acc += v * v;
    }
    part[threadIdx.x] = acc;
    __syncthreads();

    if (threadIdx.x < S) {
        float tot = 0.0f;
        for (int gg = 0; gg < 64; ++gg) tot += part[gg * S + threadIdx.x];
        colv[threadIdx.x] = powf(tot, 0.05f);  // (sqrt(tot))^(1/10)
    }
    __syncthreads();

    if (threadIdx.x == 0) {
        float m = colv[0];
        for (int t = 1; t < S; ++t) m = fmaxf(m, colv[t]);
        vals[b] = m;
    }
}

// ---------------------------------------------------------------------------
// Kernel 5: mean over batches -> scalar output.
// ---------------------------------------------------------------------------
__global__ void finalize_mean(const float* __restrict__ vals,
                              float* __restrict__ out) {
    if (blockIdx.x == 0 && threadIdx.x == 0) {
        float sum = 0.0f;
        for (int i = 0; i < BATCH; ++i) sum += vals[i];
        out[0] = sum / (float)BATCH;
    }
}

// ---------------------------------------------------------------------------
// Host-side launch: deterministic, graph-capture safe.
// Workspace (floats): [0,16) inv_norm | [16,48) vals |
//                     [64, 64+524288) Y | [+524288) T   (~4.2 MB)
// ---------------------------------------------------------------------------
extern "C" void kernel_launch(void* const* d_in, const int* in_sizes, int n_in,
                              void* d_out, int out_size, void* d_ws, size_t ws_size,
                              hipStream_t stream) {
    (void)in_sizes; (void)n_in; (void)out_size; (void)ws_size;
    const float* X  = (const float*)d_in[0];   // x     (32, 1024)
    const float* Cc = (const float*)d_in[1];   // circ  (32, 1024)
    const float* SP = (const float*)d_in[2];   // sphere_points (1024, 16)

    float* ws       = (float*)d_ws;
    float* inv_norm = ws;
    float* vals     = ws + 16;
    float* Y        = ws + 64;
    float* T        = Y + BATCH * PANEL;

    col_norms<<<S, 256, 0, stream>>>(SP, inv_norm);
    init_y<<<(BATCH * PANEL) / 256, 256, 0, stream>>>(SP, inv_norm, Y);

    dim3 grid(L / 16 / 8, BATCH);              // (8 tile-groups, 32 batches)
    for (int it = 0; it < N_ITERS; ++it) {
        toeplitz_mm<false><<<grid, 256, 0, stream>>>(X,  Y, T);  // T = A @ Y
        toeplitz_mm<true ><<<grid, 256, 0, stream>>>(Cc, T, Y);  // Y -= Ct @ T
    }

    batch_val<<<BATCH, 1024, 0, stream>>>(Y, SP, inv_norm, vals);
    finalize_mean<<<1, 32, 0, stream>>>(vals, (float*)d_out);
}